// SEBlock_58067957842627
// MI455X (gfx1250) — compile-verified
//
#include <hip/hip_runtime.h>
#include <math.h>

typedef float v2f  __attribute__((ext_vector_type(2)));
typedef float v8f  __attribute__((ext_vector_type(8)));
typedef float fv4  __attribute__((ext_vector_type(4)));

#define HW    3136   // 56*56
#define HW4   784    // HW/4
#define CCH   256
#define CR    64
#define BATCH 32

// ---------------------------------------------------------------------------
// Kernel 1: squeeze — per-(b,c) mean over the 56x56 plane.
// One 256-thread block per plane; b128 loads; LDS tree reduction.
// Default (RT) temporal hint deliberately pulls x into the 192MB L2 so the
// scale pass can re-read it from L2 instead of HBM.
// ---------------------------------------------------------------------------
__global__ __launch_bounds__(256) void se_squeeze(const float* __restrict__ x,
                                                  float* __restrict__ means) {
    __shared__ float red[256];
    const int p = blockIdx.x;                 // plane = b*256 + c
    const int t = threadIdx.x;
    const fv4* xp = (const fv4*)(x + (size_t)p * HW);
    float s = 0.0f;
    for (int j = t; j < HW4; j += 256) {
        fv4 v = xp[j];
        s += (v.x + v.y) + (v.z + v.w);
    }
    red[t] = s;
    __syncthreads();
    for (int off = 128; off > 0; off >>= 1) {
        if (t < off) red[t] += red[t + off];
        __syncthreads();
    }
    if (t == 0) means[p] = red[0] * (1.0f / (float)HW);
}

// ---------------------------------------------------------------------------
// Kernel 2: excite — two tiny fp32 GEMMs on V_WMMA_F32_16X16X4_F32.
//   GEMM1: [32,256] x w1^T[256,64]  -> relu(+b1) -> h[32,64] (LDS)
//   GEMM2: [32, 64] x w2^T[ 64,256] -> sigmoid(+b2) -> scale[32,256]
// 1 block x 256 threads = 8 wave32s. GEMM1: 8 tiles (1/wave). GEMM2: 32 tiles
// (4/wave). Fragment layouts per CDNA5 ISA §7.12.2:
//   A(16x4,f32)/B(4x16,f32): lane L holds K = 2*(L>>4)+{0,1} at row/col L&15
//   C/D(16x16,f32): vgpr i, lane L -> M = i + 8*(L>>4), N = L&15
// EXEC is all-1s throughout (uniform control flow), as WMMA requires.
// ---------------------------------------------------------------------------
__global__ __launch_bounds__(256) void se_excite(const float* __restrict__ means,
                                                 const float* __restrict__ w1,
                                                 const float* __restrict__ b1,
                                                 const float* __restrict__ w2,
                                                 const float* __restrict__ b2,
                                                 float* __restrict__ scale) {
    __shared__ float sse[BATCH * CCH];  // squeezed input, 32x256
    __shared__ float sh[BATCH * CR];    // hidden relu output, 32x64

    const int t = threadIdx.x;
    for (int i = t; i < BATCH * CCH; i += 256) sse[i] = means[i];
    __syncthreads();

    const int lane  = t & 31;
    const int wave  = t >> 5;
    const int lid   = lane & 15;   // row of A / col of B / N of C
    const int khalf = lane >> 4;   // 0: K pair {0,1}; 1: K pair {2,3}

    // ---- GEMM1: D[32,64] = se[32,256] * w1^T ----
    {
        const int m0 = (wave & 1) * 16;   // batch tile
        const int n0 = (wave >> 1) * 16;  // out-channel tile (0..3)
        v8f acc = {};
        for (int kk = 0; kk < CCH; kk += 4) {
            const int k = kk + 2 * khalf;
            v2f a, b;
            a.x = sse[(m0 + lid) * CCH + k];
            a.y = sse[(m0 + lid) * CCH + k + 1];
            b.x = w1[(n0 + lid) * CCH + k];      // B[k][n] = w1[n][k]
            b.y = w1[(n0 + lid) * CCH + k + 1];
            acc = __builtin_amdgcn_wmma_f32_16x16x4_f32(
                false, a, false, b, (short)0, acc, false, false);
        }
        const int n = n0 + lid;
        const float bias = b1[n];
#pragma unroll
        for (int i = 0; i < 8; ++i) {
            const int mm = m0 + i + 8 * khalf;
            const float v = acc[i] + bias;
            sh[mm * CR + n] = v > 0.0f ? v : 0.0f;
        }
    }
    __syncthreads();

    // ---- GEMM2: D[32,256] = h[32,64] * w2^T ----
    for (int tt = 0; tt < 4; ++tt) {
        const int tile = wave * 4 + tt;    // 0..31
        const int m0 = (tile & 1) * 16;
        const int n0 = (tile >> 1) * 16;   // 0..15
        v8f acc = {};
        for (int kk = 0; kk < CR; kk += 4) {
            const int k = kk + 2 * khalf;
            v2f a, b;
            a.x = sh[(m0 + lid) * CR + k];
            a.y = sh[(m0 + lid) * CR + k + 1];
            b.x = w2[(n0 + lid) * CR + k];       // B[k][n] = w2[n][k]
            b.y = w2[(n0 + lid) * CR + k + 1];
            acc = __builtin_amdgcn_wmma_f32_16x16x4_f32(
                false, a, false, b, (short)0, acc, false, false);
        }
        const int n = n0 + lid;
        const float bias = b2[n];
#pragma unroll
        for (int i = 0; i < 8; ++i) {
            const int mm = m0 + i + 8 * khalf;
            const float v = acc[i] + bias;
            scale[mm * CCH + n] = 1.0f / (1.0f + __expf(-v));
        }
    }
}

// ---------------------------------------------------------------------------
// Kernel 3: scale — out = x * scale[b,c], b128 loads, non-temporal b128
// stores so the output stream doesn't evict x from L2 while we re-read it.
// Planes are 3136 floats (= 784 float4, 16B-aligned), so a float4 never
// straddles a channel boundary.
// ---------------------------------------------------------------------------
__global__ __launch_bounds__(256) void se_scale(const float* __restrict__ x,
                                                const float* __restrict__ scale,
                                                float* __restrict__ out) {
    const int gid   = blockIdx.x * 256 + threadIdx.x;  // float4 index
    const int plane = gid / HW4;                       // b*256 + c
    const float s   = scale[plane];
    fv4 v = ((const fv4*)x)[gid];
    fv4 o;
    o.x = v.x * s; o.y = v.y * s; o.z = v.z * s; o.w = v.w * s;
    __builtin_nontemporal_store(o, ((fv4*)out) + gid);
}

// ---------------------------------------------------------------------------
extern "C" void kernel_launch(void* const* d_in, const int* in_sizes, int n_in,
                              void* d_out, int out_size, void* d_ws, size_t ws_size,
                              hipStream_t stream) {
    const float* x  = (const float*)d_in[0];
    const float* w1 = (const float*)d_in[1];
    const float* b1 = (const float*)d_in[2];
    const float* w2 = (const float*)d_in[3];
    const float* b2 = (const float*)d_in[4];
    float* out = (float*)d_out;

    float* means = (float*)d_ws;                 // [32*256] fp32
    float* scale = means + BATCH * CCH;          // [32*256] fp32

    se_squeeze<<<BATCH * CCH, 256, 0, stream>>>(x, means);
    se_excite<<<1, 256, 0, stream>>>(means, w1, b1, w2, b2, scale);

    const int n4 = (BATCH * CCH * HW) / 4;       // 6,422,528 float4
    se_scale<<<n4 / 256, 256, 0, stream>>>(x, scale, out);
}